// Loss_50233937494630
// MI455X (gfx1250) — compile-verified
//
#include <hip/hip_runtime.h>
#include <hip/hip_bf16.h>

typedef __attribute__((ext_vector_type(2))) float v2f;
typedef __attribute__((ext_vector_type(8))) float v8f;

// ---------------------------------------------------------------------------
// init: zero the scalar accumulator (harness poisons d_out with 0xAA)
// ---------------------------------------------------------------------------
__global__ void init_kernel(float* out) {
    if (threadIdx.x == 0 && blockIdx.x == 0) out[0] = 0.0f;
}

// ---------------------------------------------------------------------------
// kNN: per point, 9 smallest squared distances (incl self), write idx[1..8].
// LDS-tiled candidate streaming; branchless sorted insert.
// Each block (256 threads) lies entirely within one batch (N % 256 == 0).
// ---------------------------------------------------------------------------
__global__ __launch_bounds__(256) void knn_kernel(
    const float* __restrict__ sp,  // [B,N,3]
    int* __restrict__ idx,         // [B,N,8] out
    int Nn)
{
    const int gid = blockIdx.x * blockDim.x + threadIdx.x;
    const int b = gid / Nn;
    const int n = gid % Nn;
    const float* base = sp + (size_t)b * Nn * 3;
    const float px = base[n * 3 + 0];
    const float py = base[n * 3 + 1];
    const float pz = base[n * 3 + 2];

    float bd[9];
    int   bi[9];
#pragma unroll
    for (int k = 0; k < 9; ++k) { bd[k] = 3.0e38f; bi[k] = 0; }

    __shared__ float sx[256], sy[256], sz[256];
    for (int jt = 0; jt < Nn; jt += 256) {
        const int j = jt + threadIdx.x;
        sx[threadIdx.x] = base[j * 3 + 0];
        sy[threadIdx.x] = base[j * 3 + 1];
        sz[threadIdx.x] = base[j * 3 + 2];
        __syncthreads();
        for (int jj = 0; jj < 256; ++jj) {
            const float dx = px - sx[jj];
            const float dy = py - sy[jj];
            const float dz = pz - sz[jj];
            const float d = dx * dx + dy * dy + dz * dz;
            if (d < bd[8]) {
                float dd = d;
                int   ji = jt + jj;
#pragma unroll
                for (int k = 0; k < 9; ++k) {
                    const bool less = dd < bd[k];
                    const float td = bd[k];
                    const int   ti = bi[k];
                    bd[k] = less ? dd : td;
                    bi[k] = less ? ji : ti;
                    dd = less ? td : dd;
                    ji = less ? ti : ji;
                }
            }
        }
        __syncthreads();
    }
    int* op = idx + (size_t)gid * 8;
#pragma unroll
    for (int k = 0; k < 8; ++k) op[k] = bi[k + 1];  // drop self
}

// ---------------------------------------------------------------------------
// ARAP: sum over t of ||sqrt(|dn|^2+eps) - sqrt(|sn|^2+eps)||^2, / B
// ---------------------------------------------------------------------------
__global__ __launch_bounds__(256) void arap_kernel(
    const float* __restrict__ dp_all,  // [T,B,N,3]
    const float* __restrict__ sp,      // [B,N,3]
    const int* __restrict__ idx,       // [B,N,8]
    float* __restrict__ out,
    int Tn, int Bn, int Nn, float invB)
{
    const int gid = blockIdx.x * blockDim.x + threadIdx.x;
    const int b = gid / Nn;
    const int n = gid % Nn;

    const float* sb = sp + (size_t)b * Nn * 3;
    const float sx0 = sb[n * 3 + 0];
    const float sy0 = sb[n * 3 + 1];
    const float sz0 = sb[n * 3 + 2];

    int nb[8];
#pragma unroll
    for (int k = 0; k < 8; ++k) nb[k] = idx[(size_t)gid * 8 + k];

    float sd[8];
#pragma unroll
    for (int k = 0; k < 8; ++k) {
        const int j = nb[k];
        const float dx = sb[j * 3 + 0] - sx0;
        const float dy = sb[j * 3 + 1] - sy0;
        const float dz = sb[j * 3 + 2] - sz0;
        sd[k] = sqrtf(dx * dx + dy * dy + dz * dz + 1e-5f);
    }

    float acc = 0.0f;
    for (int t = 0; t < Tn; ++t) {
        const float* db = dp_all + ((size_t)t * Bn + b) * Nn * 3;
        const float dx0 = db[n * 3 + 0];
        const float dy0 = db[n * 3 + 1];
        const float dz0 = db[n * 3 + 2];
#pragma unroll
        for (int k = 0; k < 8; ++k) {
            const int j = nb[k];
            const float dx = db[j * 3 + 0] - dx0;
            const float dy = db[j * 3 + 1] - dy0;
            const float dz = db[j * 3 + 2] - dz0;
            const float dd = sqrtf(dx * dx + dy * dy + dz * dz + 1e-5f);
            const float df = dd - sd[k];
            acc += df * df;
        }
    }

    __shared__ float red[256];
    red[threadIdx.x] = acc;
    __syncthreads();
    for (int s = 128; s > 0; s >>= 1) {
        if (threadIdx.x < s) red[threadIdx.x] += red[threadIdx.x + s];
        __syncthreads();
    }
    if (threadIdx.x == 0) atomicAdd(out, red[0] * invB);
}

// ---------------------------------------------------------------------------
// Chamfer via V_WMMA_F32_16X16X4_F32.
// One wave owns one 16-row tile of x for a given (t, b, direction); it loops
// over all 16-col tiles of y, computing inner products via WMMA and tracking
// the per-row running min of d = |x|^2 + |y|^2 - 2 x.y.
//   A layout (16x4 f32): lanes 0-15 -> (K=0,K=1)=(x,y) of row l; lanes 16-31
//   -> (K=2,K=3)=(z,0) of row l-16. B layout mirrors it (col = l&15).
//   C/D layout: VGPR v holds row v (lanes 0-15) / row v+8 (lanes 16-31),
//   column = lane & 15.
// All loads are unconditional (addresses always in-bounds); the K=3 zero pad
// is a post-load v_cndmask so the inner loop has no EXEC manipulation.
// dir==0: rows=dp[t], cols=target (-> d1). dir==1: swapped (-> d2).
// ---------------------------------------------------------------------------
__global__ __launch_bounds__(128) void chamfer_min_kernel(
    const float* __restrict__ dp_all,  // [T,B,N,3]
    const float* __restrict__ tgt,     // [B,N,3]
    float* __restrict__ out,
    int Bn, int Nn, float scale)       // scale = 0.5/B
{
    const int t    = blockIdx.y;
    const int dir  = blockIdx.z;
    const int widx = threadIdx.x >> 5;
    const int lane = threadIdx.x & 31;
    const int wglobal = blockIdx.x * (blockDim.x >> 5) + widx;
    const int tilesPerB = Nn / 16;
    const int b = wglobal / tilesPerB;
    const int mtile = wglobal % tilesPerB;
    if (b >= Bn) return;

    const float* dp = dp_all + ((size_t)t * Bn + b) * Nn * 3;
    const float* tg = tgt + (size_t)b * Nn * 3;
    const float* xp = dir ? tg : dp;  // row matrix
    const float* yp = dir ? dp : tg;  // col matrix

    const int hf   = lane >> 4;      // 0: holds K=0,1; 1: holds K=2,pad
    const int l15  = lane & 15;
    const int m0   = mtile * 16;
    const int off0 = hf << 1;        // 0 -> read x, 2 -> read z

    // A matrix: 16 rows x 4 (xyz + zero pad); unconditional loads + cndmask
    const float* xr = xp + (size_t)(m0 + l15) * 3;
    v2f a;
    {
        const float l0 = xr[off0];   // x (half0) or z (half1)
        const float l1 = xr[1];      // y (always in-bounds)
        a.x = l0;
        a.y = hf ? 0.0f : l1;
    }

    // |x|^2 per row: combine the two lane-halves, then broadcast per-VGPR row
    float p = a.x * a.x + a.y * a.y;
    const float x2full = p + __shfl_xor(p, 16, 32);  // lane L: |x_{m0+(L&15)}|^2
    float x2m[8];
#pragma unroll
    for (int v = 0; v < 8; ++v)
        x2m[v] = __shfl(x2full, (hf << 3) + v, 32);  // row v (half0) / v+8 (half1)

    float rowmin[8];
#pragma unroll
    for (int v = 0; v < 8; ++v) rowmin[v] = 3.0e38f;

    const v8f czero = {};
    const float* yr = yp + (size_t)l15 * 3;
    for (int jt = 0; jt < tilesPerB; ++jt, yr += 48) {
        v2f bb;
        const float l0 = yr[off0];   // unconditional
        const float l1 = yr[1];      // unconditional
        bb.x = l0;
        bb.y = hf ? 0.0f : l1;

        float q = bb.x * bb.x + bb.y * bb.y;
        const float y2col = q + __shfl_xor(q, 16, 32);  // |y_{n0+(lane&15)}|^2

        // inner[m][n] tile:  emits v_wmma_f32_16x16x4_f32
        v8f c = __builtin_amdgcn_wmma_f32_16x16x4_f32(
            false, a, false, bb, (short)0, czero, false, false);

#pragma unroll
        for (int v = 0; v < 8; ++v) {
            const float d = x2m[v] + y2col - 2.0f * c[v];
            rowmin[v] = fminf(rowmin[v], d);
        }
    }

    // min-reduce across the 16 lanes of each half (rows stay in their half)
#pragma unroll
    for (int off = 1; off < 16; off <<= 1) {
#pragma unroll
        for (int v = 0; v < 8; ++v)
            rowmin[v] = fminf(rowmin[v], __shfl_xor(rowmin[v], off, 32));
    }
    if (l15 == 0) {  // lane 0: rows m0..m0+7, lane 16: rows m0+8..m0+15
        float s = 0.0f;
#pragma unroll
        for (int v = 0; v < 8; ++v) s += rowmin[v];
        atomicAdd(out, s * scale);
    }
}

// ---------------------------------------------------------------------------
// misc: pd (||drp-dp||^2 / B) + sparse (mean |pw|) + tran (masked R frob / B)
// ---------------------------------------------------------------------------
__global__ __launch_bounds__(256) void misc_kernel(
    const float* __restrict__ pw,    // [T,B,N,1]
    const float* __restrict__ drp,   // [T,B,N,3]
    const float* __restrict__ dpa,   // [T,B,N,3]
    const float* __restrict__ rm,    // [T,B,4,4]
    float* __restrict__ out,
    int pd_n, int sp_n, int tran_n, float invB, float invBN)
{
    const int i = blockIdx.x * blockDim.x + threadIdx.x;
    float acc = 0.0f;
    if (i < pd_n) {
        const float d = drp[i] - dpa[i];
        acc += d * d * invB;
    }
    if (i < sp_n) acc += fabsf(pw[i]) * invBN;
    if (i < tran_n) {
        const int r  = i % 3;       // row 0..2, column 3
        const int tb = i / 3;       // t*B + b
        const float v = rm[(size_t)tb * 16 + r * 4 + 3];
        acc += v * v * invB;
    }
    __shared__ float red[256];
    red[threadIdx.x] = acc;
    __syncthreads();
    for (int s = 128; s > 0; s >>= 1) {
        if (threadIdx.x < s) red[threadIdx.x] += red[threadIdx.x + s];
        __syncthreads();
    }
    if (threadIdx.x == 0) atomicAdd(out, red[0]);
}

// ---------------------------------------------------------------------------
extern "C" void kernel_launch(void* const* d_in, const int* in_sizes, int n_in,
                              void* d_out, int out_size, void* d_ws, size_t ws_size,
                              hipStream_t stream) {
    // setup_inputs order:
    // 0 phi [T,B,1024] (unused by loss)
    // 1 point_weight [T,B,N,1]
    // 2 deform_rigid_points [T,B,N,3]
    // 3 deformation_points [T,B,N,3]
    // 4 rigid_matrix [T,B,4,4]
    // 5 source_points [B,N,3]
    // 6 target_points [B,N,3]
    const float* pw  = (const float*)d_in[1];
    const float* drp = (const float*)d_in[2];
    const float* dpa = (const float*)d_in[3];
    const float* rm  = (const float*)d_in[4];
    const float* sp  = (const float*)d_in[5];
    const float* tg  = (const float*)d_in[6];
    float* out = (float*)d_out;
    int*   idx = (int*)d_ws;          // B*N*8 ints = 512 KB

    const int T = 3, B = 4, N = 4096;

    init_kernel<<<1, 1, 0, stream>>>(out);

    knn_kernel<<<(B * N) / 256, 256, 0, stream>>>(sp, idx, N);

    arap_kernel<<<(B * N) / 256, 256, 0, stream>>>(dpa, sp, idx, out,
                                                   T, B, N, 1.0f / B);

    {   // 4 waves per block; grid.y = t, grid.z = direction
        dim3 grid((B * (N / 16)) / 4, T, 2);
        chamfer_min_kernel<<<grid, 128, 0, stream>>>(dpa, tg, out, B, N,
                                                     0.5f / B);
    }

    {
        const int pd_n   = T * B * N * 3;
        const int sp_n   = T * B * N;
        const int tran_n = T * B * 3;
        misc_kernel<<<(pd_n + 255) / 256, 256, 0, stream>>>(
            pw, drp, dpa, rm, out, pd_n, sp_n, tran_n,
            1.0f / B, 1.0f / (B * N));
    }
}